// BatchNeuralKB_81346680586349
// MI455X (gfx1250) — compile-verified
//
#include <hip/hip_runtime.h>
#include <math.h>

typedef __attribute__((ext_vector_type(2))) float v2f;
typedef __attribute__((ext_vector_type(8))) float v8f;

#define B_  128
#define F_  2048
#define E_  256
#define CE_ 768   // 3*E concatenated

// ---------------------------------------------------------------------------
// init: seed per-batch min(sq_dist) workspace with +INF (as uint bits).
// ---------------------------------------------------------------------------
__global__ void kb_init_ws(unsigned* __restrict__ ws) {
    ws[threadIdx.x] = 0x7F800000u;  // +inf
}

// ---------------------------------------------------------------------------
// main: each wave owns a 16-fact tile of one batch row.
//   G = facts_tile - q  (16 x 768), sq_dist = diag(G * G^T)
//   accumulated with V_WMMA_F32_16X16X4_F32 (A-frag == B-frag for G*G^T).
// ---------------------------------------------------------------------------
__global__ __launch_bounds__(256) void kb_min_dist(
    const float* __restrict__ rel,  const float* __restrict__ arg1,
    const float* __restrict__ arg2, const float* __restrict__ frel,
    const float* __restrict__ farg1, const float* __restrict__ farg2,
    const int* __restrict__ nb_facts, unsigned* __restrict__ ws)
{
    __shared__ float qs[CE_];

    const int b = blockIdx.y;

    // Stage concatenated query [rel|arg1|arg2] into LDS (256 threads x 3).
    {
        const int t = threadIdx.x;
        qs[t]          = rel [b * E_ + t];
        qs[t + E_]     = arg1[b * E_ + t];
        qs[t + 2 * E_] = arg2[b * E_ + t];
    }
    __syncthreads();

    const int lane  = threadIdx.x & 31;
    const int wave  = threadIdx.x >> 5;
    const int m     = lane & 15;       // fact-row within tile (A-matrix M)
    const int hi    = lane >> 4;       // K-half select per WMMA f32 layout
    const int fbase = (blockIdx.x * 8 + wave) * 16;

    v8f c = {0.f, 0.f, 0.f, 0.f, 0.f, 0.f, 0.f, 0.f};

    const float* fptrs[3] = {frel, farg1, farg2};
    #pragma unroll
    for (int t = 0; t < 3; ++t) {
        // This lane streams row (fbase+m) of tensor t, starting at k-offset 2*hi.
        const float* fp = fptrs[t] + (((size_t)b * F_ + (size_t)(fbase + m)) * E_) + 2 * hi;
        const float* qp = qs + t * E_ + 2 * hi;
        #pragma unroll 16
        for (int kc = 0; kc < E_ / 4; ++kc) {
            v2f fv = *(const v2f*)(fp + kc * 4);   // global_load_b64, 8B aligned
            v2f qv = *(const v2f*)(qp + kc * 4);   // ds_load_b64
            v2f g  = fv - qv;
            // D = G_chunk * G_chunk^T + C ; A-frag and B-frag coincide for G*G^T.
            c = __builtin_amdgcn_wmma_f32_16x16x4_f32(
                    /*neg_a=*/false, g, /*neg_b=*/false, g,
                    /*c_mod=*/(short)0, c, /*reuse_a=*/false, /*reuse_b=*/false);
        }
    }

    // Extract diag(c): D[M][N], N = lane&15, M = vgpr + 8*hi.
    // Lane holds diag element iff (m >> 3) == hi, at vgpr index m & 7.
    float d = __builtin_inff();
    if ((m >> 3) == hi) {
        const int v = m & 7;
        float dv = c[0];
        #pragma unroll
        for (int i = 1; i < 8; ++i)
            if (v == i) dv = c[i];
        const int nb = nb_facts[b];
        if (fbase + m < nb) d = dv;   // mask padded facts -> +inf
    }

    // wave32 min-reduce
    #pragma unroll
    for (int off = 16; off > 0; off >>= 1)
        d = fminf(d, __shfl_xor(d, off, 32));

    // sq_dist >= 0 (or +inf): IEEE bits order == unsigned order.
    if (lane == 0)
        atomicMin(ws + b, __float_as_uint(d));
}

// ---------------------------------------------------------------------------
// finalize: out[b] = nb>0 ? exp(-min_dist) : 0
// ---------------------------------------------------------------------------
__global__ void kb_finalize(const unsigned* __restrict__ ws,
                            const int* __restrict__ nb_facts,
                            float* __restrict__ out)
{
    const int b = threadIdx.x;
    const float dmin = __uint_as_float(ws[b]);
    out[b] = (nb_facts[b] > 0) ? expf(-dmin) : 0.0f;
}

// ---------------------------------------------------------------------------
extern "C" void kernel_launch(void* const* d_in, const int* in_sizes, int n_in,
                              void* d_out, int out_size, void* d_ws, size_t ws_size,
                              hipStream_t stream)
{
    const float* rel   = (const float*)d_in[0];
    const float* arg1  = (const float*)d_in[1];
    const float* arg2  = (const float*)d_in[2];
    const float* frel  = (const float*)d_in[3];
    const float* farg1 = (const float*)d_in[4];
    const float* farg2 = (const float*)d_in[5];
    const int*   nb    = (const int*)d_in[6];

    unsigned* ws = (unsigned*)d_ws;

    kb_init_ws<<<1, B_, 0, stream>>>(ws);

    dim3 grid(F_ / 128, B_);   // 16 tiles-of-8-waves x 128 batches
    kb_min_dist<<<grid, 256, 0, stream>>>(rel, arg1, arg2, frel, farg1, farg2, nb, ws);

    kb_finalize<<<1, B_, 0, stream>>>(ws, nb, (float*)d_out);
}